// GAT_66185446031494
// MI455X (gfx1250) — compile-verified
//
#include <hip/hip_runtime.h>
#include <math.h>

// GAT on gfx1250: f32 WMMA strip-GEMMs + CSR-based edge softmax aggregation.

typedef __attribute__((ext_vector_type(2))) float v2f;
typedef __attribute__((ext_vector_type(8))) float v8f;

#define LRELU_SLOPE 0.2f

// ---------------------------------------------------------------- utilities
__global__ void zero2_int_kernel(int* __restrict__ a, int* __restrict__ b, int n) {
  int i = blockIdx.x * blockDim.x + threadIdx.x;
  if (i < n) { a[i] = 0; b[i] = 0; }
}

__global__ void hist_kernel(const int* __restrict__ dst, int* __restrict__ cnt, int e) {
  int i = blockIdx.x * blockDim.x + threadIdx.x;
  if (i < e) atomicAdd(&cnt[dst[i]], 1);
}

// single-workgroup inclusive scan -> row_ptr (exclusive at [i], total at [n])
__global__ void scan_kernel(const int* __restrict__ cnt, int* __restrict__ row_ptr, int n) {
  __shared__ int sdata[1024];
  __shared__ int s_running;
  const int tid = threadIdx.x;
  if (tid == 0) { s_running = 0; row_ptr[0] = 0; }
  __syncthreads();
  for (int base = 0; base < n; base += 1024) {
    int i = base + tid;
    int v = (i < n) ? cnt[i] : 0;
    sdata[tid] = v;
    __syncthreads();
    for (int off = 1; off < 1024; off <<= 1) {
      int t = (tid >= off) ? sdata[tid - off] : 0;
      __syncthreads();
      sdata[tid] += t;
      __syncthreads();
    }
    if (i < n) row_ptr[i + 1] = s_running + sdata[tid];
    __syncthreads();
    if (tid == 0) s_running += sdata[1023];
    __syncthreads();
  }
}

__global__ void fill_kernel(const int* __restrict__ src, const int* __restrict__ dst,
                            const float* __restrict__ w, const int* __restrict__ row_ptr,
                            int* __restrict__ cursor, int* __restrict__ es,
                            float* __restrict__ ew, int e) {
  int i = blockIdx.x * blockDim.x + threadIdx.x;
  if (i >= e) return;
  int d = dst[i];
  int pos = atomicAdd(&cursor[d], 1);
  int idx = row_ptr[d] + pos;
  es[idx] = src[i];
  ew[idx] = w[i];
}

// ---------------------------------------------------------------- WMMA GEMM
// Strip GEMM: C[M, NT*16] = A[M,K] * B[K, NT*16], row-major.
// One wave owns a 16-row strip and ALL NT column tiles (register
// accumulators), so each A fragment is loaded once and reused NT times.
// Per k-step: batch-load all NT B fragments into distinct registers (one
// load clause), then issue NT back-to-back WMMAs — avoids the
// load->wait(0)->wmma serialization seen when B registers are recycled.
// Fragment layout per CDNA5 ISA 7.12.2:
//   A 16x4 f32 : lanes<16 -> rows, {K=k0,k0+1}; lanes>=16 -> {K=k0+2,k0+3}
//   B  4x16 f32: lanes<16 -> cols, {K=k0,k0+1}; lanes>=16 -> {K=k0+2,k0+3}
//   C 16x16 f32: vgpr i = (M = 8*(lane>>4)+i, N = lane&15)
template <int NT>
__global__ void gemm_strip_wmma_kernel(const float* __restrict__ A,
                                       const float* __restrict__ B,
                                       float* __restrict__ C, int M, int K) {
  const int Nn = NT * 16;
  const int wave = (blockIdx.x * blockDim.x + threadIdx.x) >> 5;
  const int lane = threadIdx.x & 31;
  if (wave * 16 >= M) return;  // wave-uniform: EXEC stays all-ones for WMMA

  const int row = wave * 16 + (lane & 15);
  const int col = lane & 15;
  const int khalf = (lane >> 4) << 1;

  v8f acc[NT];
#pragma unroll
  for (int t = 0; t < NT; ++t) acc[t] = (v8f){};

  for (int k0 = 0; k0 < K; k0 += 4) {
    v2f a;
    a.x = A[(size_t)row * K + k0 + khalf];
    a.y = A[(size_t)row * K + k0 + khalf + 1];
    const float* __restrict__ b0 = B + (size_t)(k0 + khalf) * Nn + col;
    v2f b[NT];
#pragma unroll
    for (int t = 0; t < NT; ++t) {       // one clause of 2*NT loads
      b[t].x = b0[t * 16];
      b[t].y = b0[Nn + t * 16];
    }
#pragma unroll
    for (int t = 0; t < NT; ++t) {       // back-to-back WMMAs, shared A
      acc[t] = __builtin_amdgcn_wmma_f32_16x16x4_f32(false, a, false, b[t],
                                                     (short)0, acc[t], false, false);
    }
  }
  const int baseM = wave * 16 + ((lane >> 4) << 3);
#pragma unroll
  for (int t = 0; t < NT; ++t) {
#pragma unroll
    for (int i = 0; i < 8; ++i)
      C[(size_t)(baseM + i) * Nn + t * 16 + col] = acc[t][i];
  }
}

// ---------------------------------------------------------------- attn logits
// el[n,h] = <h[n,h,:], al[h,:]> ; er likewise. One thread per (node, head).
__global__ void logits_kernel(const float* __restrict__ h, const float* __restrict__ al,
                              const float* __restrict__ ar, float* __restrict__ el,
                              float* __restrict__ er, int n, int heads, int d) {
  int idx = blockIdx.x * blockDim.x + threadIdx.x;
  if (idx >= n * heads) return;
  int hh = idx % heads;
  const float* __restrict__ hp = h + (size_t)idx * d;
  const float* __restrict__ alp = al + (size_t)hh * d;
  const float* __restrict__ arp = ar + (size_t)hh * d;
  float sl = 0.f, sr = 0.f;
  for (int j = 0; j < d; ++j) { float v = hp[j]; sl += v * alp[j]; sr += v * arp[j]; }
  el[idx] = sl;
  er[idx] = sr;
}

// ---------------------------------------------------------------- aggregation
// Layer 1: H=8, D=16, width 128 -> 4 floats/lane. out = relu(sum/z + b).
__global__ void aggregate1_kernel(const float* __restrict__ h1, const float* __restrict__ el,
                                  const float* __restrict__ er, const int* __restrict__ row_ptr,
                                  const int* __restrict__ es, const float* __restrict__ ew,
                                  const float* __restrict__ bias, float* __restrict__ x2, int n) {
  const int wid = (blockIdx.x * blockDim.x + threadIdx.x) >> 5;
  const int lane = threadIdx.x & 31;
  if (wid >= n) return;
  const int head = lane >> 2;     // 4 lanes per head (D=16 / 4 floats)
  const int off = lane << 2;      // feature offset within the 128-wide row
  const int s0 = row_ptr[wid], s1 = row_ptr[wid + 1];
  const float ern = er[wid * 8 + head];

  // pass 1: per-head max (stable softmax), prefetch feature rows for pass 2
  float m = -3.402823466e38f;
  for (int p = s0; p < s1; ++p) {
    int s = es[p];
    float ev = el[s * 8 + head] + ern;
    ev = ev > 0.f ? ev : LRELU_SLOPE * ev;
    m = fmaxf(m, ev);
    __builtin_prefetch(h1 + (size_t)s * 128 + off, 0, 0);
  }
  // pass 2: z-sum and weighted feature accumulation in registers
  float z = 0.f;
  float4 acc = make_float4(0.f, 0.f, 0.f, 0.f);
  for (int p = s0; p < s1; ++p) {
    int s = es[p];
    float ev = el[s * 8 + head] + ern;
    ev = ev > 0.f ? ev : LRELU_SLOPE * ev;
    float a = __expf(ev - m);
    z += a;
    float c = a * ew[p];
    float4 hv = *(const float4*)(h1 + (size_t)s * 128 + off);
    acc.x += c * hv.x; acc.y += c * hv.y; acc.z += c * hv.z; acc.w += c * hv.w;
  }
  const float inv = (z > 0.f) ? (1.f / z) : 0.f;  // deg-0 node -> relu(bias)
  float4 bv = *(const float4*)(bias + off);
  float4 o;
  o.x = fmaxf(acc.x * inv + bv.x, 0.f);
  o.y = fmaxf(acc.y * inv + bv.y, 0.f);
  o.z = fmaxf(acc.z * inv + bv.z, 0.f);
  o.w = fmaxf(acc.w * inv + bv.w, 0.f);
  *(float4*)(x2 + (size_t)wid * 128 + off) = o;
}

// Layer 2: H=1, D=64 -> 2 floats/lane. out = sum/z + b (no activation).
__global__ void aggregate2_kernel(const float* __restrict__ h2, const float* __restrict__ el,
                                  const float* __restrict__ er, const int* __restrict__ row_ptr,
                                  const int* __restrict__ es, const float* __restrict__ ew,
                                  const float* __restrict__ bias, float* __restrict__ out, int n) {
  const int wid = (blockIdx.x * blockDim.x + threadIdx.x) >> 5;
  const int lane = threadIdx.x & 31;
  if (wid >= n) return;
  const int off = lane << 1;
  const int s0 = row_ptr[wid], s1 = row_ptr[wid + 1];
  const float ern = er[wid];

  float m = -3.402823466e38f;
  for (int p = s0; p < s1; ++p) {
    int s = es[p];
    float ev = el[s] + ern;
    ev = ev > 0.f ? ev : LRELU_SLOPE * ev;
    m = fmaxf(m, ev);
    __builtin_prefetch(h2 + (size_t)s * 64 + off, 0, 0);
  }
  float z = 0.f;
  float2 acc = make_float2(0.f, 0.f);
  for (int p = s0; p < s1; ++p) {
    int s = es[p];
    float ev = el[s] + ern;
    ev = ev > 0.f ? ev : LRELU_SLOPE * ev;
    float a = __expf(ev - m);
    z += a;
    float c = a * ew[p];
    float2 hv = *(const float2*)(h2 + (size_t)s * 64 + off);
    acc.x += c * hv.x; acc.y += c * hv.y;
  }
  const float inv = (z > 0.f) ? (1.f / z) : 0.f;
  float2 bv = *(const float2*)(bias + off);
  float2 o;
  o.x = acc.x * inv + bv.x;
  o.y = acc.y * inv + bv.y;
  *(float2*)(out + (size_t)wid * 64 + off) = o;
}

// ---------------------------------------------------------------- launch
extern "C" void kernel_launch(void* const* d_in, const int* in_sizes, int n_in,
                              void* d_out, int out_size, void* d_ws, size_t ws_size,
                              hipStream_t stream) {
  const float* feat = (const float*)d_in[0];
  const int*   src  = (const int*)d_in[1];
  const int*   dst  = (const int*)d_in[2];
  const float* w    = (const float*)d_in[3];
  const float* W1   = (const float*)d_in[4];
  const float* al1  = (const float*)d_in[5];
  const float* ar1  = (const float*)d_in[6];
  const float* b1   = (const float*)d_in[7];
  const float* W2   = (const float*)d_in[8];
  const float* al2  = (const float*)d_in[9];
  const float* ar2  = (const float*)d_in[10];
  const float* b2   = (const float*)d_in[11];
  float* out = (float*)d_out;

  const int n = in_sizes[0] / 128;  // nodes
  const int e = in_sizes[1];        // edges

  // carve workspace (256B-aligned slices)
  char* p = (char*)d_ws;
  auto alloc = [&](size_t bytes) -> char* {
    char* r = p;
    p += (bytes + 255) & ~(size_t)255;
    return r;
  };
  float* h1      = (float*)alloc((size_t)n * 128 * 4);
  float* x2      = (float*)alloc((size_t)n * 128 * 4);
  float* h2      = (float*)alloc((size_t)n * 64 * 4);
  float* el1     = (float*)alloc((size_t)n * 8 * 4);
  float* er1     = (float*)alloc((size_t)n * 8 * 4);
  float* el2     = (float*)alloc((size_t)n * 4);
  float* er2     = (float*)alloc((size_t)n * 4);
  int*   row_ptr = (int*)alloc((size_t)(n + 1) * 4);
  int*   cnt     = (int*)alloc((size_t)n * 4);
  int*   cursor  = (int*)alloc((size_t)n * 4);
  int*   es      = (int*)alloc((size_t)e * 4);
  float* ew      = (float*)alloc((size_t)e * 4);

  const int B = 256;

  // --- build dst-CSR (no float atomics anywhere in the pipeline)
  zero2_int_kernel<<<(n + B - 1) / B, B, 0, stream>>>(cnt, cursor, n);
  hist_kernel<<<(e + B - 1) / B, B, 0, stream>>>(dst, cnt, e);
  scan_kernel<<<1, 1024, 0, stream>>>(cnt, row_ptr, n);
  fill_kernel<<<(e + B - 1) / B, B, 0, stream>>>(src, dst, w, row_ptr, cursor, es, ew, e);

  const int strips = (n + 15) / 16;           // 16-row strips, one wave each
  const int sthreads = strips * 32;

  // --- layer 1: h1 = feat @ W1  (K=128, Nn=128 -> NT=8)
  gemm_strip_wmma_kernel<8><<<(sthreads + B - 1) / B, B, 0, stream>>>(feat, W1, h1, n, 128);
  logits_kernel<<<(n * 8 + B - 1) / B, B, 0, stream>>>(h1, al1, ar1, el1, er1, n, 8, 16);
  aggregate1_kernel<<<((n * 32) + B - 1) / B, B, 0, stream>>>(h1, el1, er1, row_ptr, es, ew,
                                                              b1, x2, n);
  // --- layer 2: h2 = x2 @ W2  (K=128, Nn=64 -> NT=4)
  gemm_strip_wmma_kernel<4><<<(sthreads + B - 1) / B, B, 0, stream>>>(x2, W2, h2, n, 128);
  logits_kernel<<<(n + B - 1) / B, B, 0, stream>>>(h2, al2, ar2, el2, er2, n, 1, 64);
  aggregate2_kernel<<<((n * 32) + B - 1) / B, B, 0, stream>>>(h2, el2, er2, row_ptr, es, ew,
                                                              b2, out, n);
  (void)n_in; (void)out_size; (void)ws_size;
}